// ModelWithConvCrf_80874234184530
// MI455X (gfx1250) — compile-verified
//
#include <hip/hip_runtime.h>
#include <hip/hip_bf16.h>
#include <math.h>

typedef __attribute__((ext_vector_type(2))) float v2f;
typedef __attribute__((ext_vector_type(8))) float v8f;

#define BN   2
#define CH   21
#define HH   512
#define WW   512
#define PH   128
#define PW   128
#define KS   11
#define K2   121
#define KC   27   // conv im2col depth (3*3*3), padded to 28 for WMMA

// ---------------------------------------------------------------------------
// Kernel 1: backbone 3x3 conv as im2col GEMM via V_WMMA_F32_16X16X4_F32.
// One wave (32 threads) computes a 16(ch) x 16(pixels) f32 tile; 7 K-steps.
// K decode is fully constant-folded (both half-wave variants computed at
// compile time, selected by `half` with v_cndmask); loads are branch-free
// (clamped address + post-load zero select) so they can clause.
// Grid: b(2) * y(512) * xtile(32) * mtile(2) = 65536 waves.
// ---------------------------------------------------------------------------
__global__ void conv_unary_wmma(const float* __restrict__ x,
                                const float* __restrict__ wgt,   // [21][27]
                                const float* __restrict__ bias,  // [21]
                                float* __restrict__ unary) {
  const int lane = threadIdx.x & 31;
  const int half = lane >> 4;      // 0: K pair {0,1}, 1: K pair {2,3}
  const int ln   = lane & 15;

  int tile = blockIdx.x;
  const int mt = tile & 1;  tile >>= 1;   // channel tile (0: ch0-15, 1: ch16-20)
  const int xt = tile & 31; tile >>= 5;   // 16-pixel column tile
  const int y  = tile & 511; tile >>= 9;  // row
  const int b  = tile;                    // batch

  const int px = xt * 16 + ln;            // this lane's N (pixel column)
  const int M  = mt * 16 + ln;            // this lane's A-row (output channel)
  const int Mc = min(M, CH - 1);
  const bool Mok = (M < CH);
  const float* __restrict__ xb = x + (size_t)b * 3 * HH * WW;

  v8f acc = {};
  #pragma unroll
  for (int ks = 0; ks < 7; ++ks) {
    v2f a, bv;
    #pragma unroll
    for (int p = 0; p < 2; ++p) {
      // k for half==0 and half==1 are compile-time constants after unroll
      const int k0 = ks * 4 + p;          // half == 0
      const int k1 = ks * 4 + 2 + p;      // half == 1
      const bool k0ok = (k0 < KC), k1ok = (k1 < KC);
      const int k0c = k0ok ? k0 : 0, k1c = k1ok ? k1 : 0;

      // ---- A: weights w[M][k], zero beyond M>=21 or k>=27 ----
      const int kk  = half ? k1c : k0c;
      const bool kok = half ? k1ok : k0ok;
      float av = wgt[Mc * KC + kk];
      av = (kok && Mok) ? av : 0.f;

      // ---- B: im2col x[ci][y+ky-1][px+kx-1], zero-padded borders ----
      const int ci = half ? (k1c / 9)       : (k0c / 9);        // constants
      const int ky = half ? ((k1c % 9) / 3) : ((k0c % 9) / 3);  // constants
      const int kx = half ? (k1c % 3)       : (k0c % 3);        // constants
      const int yy = y + ky - 1, xx = px + kx - 1;
      const bool ok = kok && (yy >= 0) && (yy < HH) && (xx >= 0) && (xx < WW);
      const int yyc = min(max(yy, 0), HH - 1);
      const int xxc = min(max(xx, 0), WW - 1);
      float bvv = xb[(ci * HH + yyc) * WW + xxc];   // always-valid address
      bvv = ok ? bvv : 0.f;

      if (p == 0) { a.x = av; bv.x = bvv; }
      else        { a.y = av; bv.y = bvv; }
    }
    acc = __builtin_amdgcn_wmma_f32_16x16x4_f32(
        /*neg_a=*/false, a, /*neg_b=*/false, bv,
        /*c_mod=*/(short)0, acc, /*reuse_a=*/false, /*reuse_b=*/false);
  }

  // D layout: lanes 0-15 -> N=ln, row r holds M=r; lanes 16-31 -> M=r+8
  #pragma unroll
  for (int r = 0; r < 8; ++r) {
    int Mo = mt * 16 + r + half * 8;
    if (Mo < CH)
      unary[((b * CH + Mo) * HH + y) * WW + px] = acc[r] + bias[Mo];
  }
}

// ---------------------------------------------------------------------------
// Kernel 2: pooled edge features (4x4 avg pool of bilateral & spatial feats).
// featB: [B,5,128,128] = {y/80, x/80, r/13, g/13, b/13}; featS: [B,2,128,128]
// ---------------------------------------------------------------------------
__global__ void pooled_feats(const float* __restrict__ x,
                             float* __restrict__ featB,
                             float* __restrict__ featS) {
  int t = blockIdx.x * blockDim.x + threadIdx.x;
  if (t >= BN * PH * PW) return;
  const int wp = t & 127, hp = (t >> 7) & 127, b = t >> 14;

  float rgb[3];
  #pragma unroll
  for (int c = 0; c < 3; ++c) {
    float s = 0.f;
    for (int dy = 0; dy < 4; ++dy)
      #pragma unroll
      for (int dx = 0; dx < 4; ++dx)
        s += x[((b * 3 + c) * HH + hp * 4 + dy) * WW + wp * 4 + dx];
    rgb[c] = s * (1.f / 16.f);
  }
  const float yc = (float)(hp * 4) + 1.5f;   // mean of rows 4hp..4hp+3
  const float xc = (float)(wp * 4) + 1.5f;

  const int base = b * 5 * (PH * PW) + hp * PW + wp;
  featB[base + 0 * PH * PW] = yc * (1.f / 80.f);
  featB[base + 1 * PH * PW] = xc * (1.f / 80.f);
  featB[base + 2 * PH * PW] = rgb[0] * (1.f / 13.f);
  featB[base + 3 * PH * PW] = rgb[1] * (1.f / 13.f);
  featB[base + 4 * PH * PW] = rgb[2] * (1.f / 13.f);

  const int bs = b * 2 * (PH * PW) + hp * PW + wp;
  featS[bs]           = yc * (1.f / 3.f);
  featS[bs + PH * PW] = xc * (1.f / 3.f);
}

// ---------------------------------------------------------------------------
// Kernel 3: pixel-adaptive Gaussian kernel (zero-padded patches: OOB -> p=0,
// so d = -center still contributes, matching conv_general_dilated_patches).
// Nested dy/dx loops: window decode folds to constants; loads clamp+select.
// ---------------------------------------------------------------------------
template <int NC>
__global__ void gauss_kernel(const float* __restrict__ feat,
                             float* __restrict__ kern) {  // [B,121,128,128]
  int t = blockIdx.x * blockDim.x + threadIdx.x;
  if (t >= BN * PH * PW) return;
  const int wp = t & 127, hp = (t >> 7) & 127, b = t >> 14;

  float ctr[NC];
  #pragma unroll
  for (int c = 0; c < NC; ++c)
    ctr[c] = feat[((b * NC + c) * PH + hp) * PW + wp];

  const int kbase = b * K2 * (PH * PW) + hp * PW + wp;
  int j = 0;
  for (int dy = -5; dy <= 5; ++dy) {
    const int yy = hp + dy;
    const bool rowok = (yy >= 0) && (yy < PH);
    const int yyc = min(max(yy, 0), PH - 1);
    #pragma unroll
    for (int dx = -5; dx <= 5; ++dx, ++j) {
      const int xx = wp + dx;
      const bool ok = rowok && (xx >= 0) && (xx < PW);
      const int xxc = min(max(xx, 0), PW - 1);
      float s = 0.f;
      #pragma unroll
      for (int c = 0; c < NC; ++c) {
        float pv = feat[((b * NC + c) * PH + yyc) * PW + xxc];
        pv = ok ? pv : 0.f;
        float d = pv - ctr[c];
        s = fmaf(d, d, s);
      }
      kern[kbase + j * (PH * PW)] = expf(-0.5f * s);
    }
  }
}

// ---------------------------------------------------------------------------
// Kernel 4: fused softmax(21ch) + 4x4 avg pool -> Qb [B,21,128,128]
// ---------------------------------------------------------------------------
__global__ void softmax_pool(const float* __restrict__ logQ,
                             float* __restrict__ Qb) {
  int t = blockIdx.x * blockDim.x + threadIdx.x;
  if (t >= BN * PH * PW) return;
  const int wp = t & 127, hp = (t >> 7) & 127, b = t >> 14;

  float acc[CH];
  #pragma unroll
  for (int c = 0; c < CH; ++c) acc[c] = 0.f;

  for (int py = 0; py < 4; ++py) {
    for (int px = 0; px < 4; ++px) {
      const int h = hp * 4 + py, w = wp * 4 + px;
      float v[CH];
      float m = -3.4e38f;
      #pragma unroll
      for (int c = 0; c < CH; ++c) {
        v[c] = logQ[((b * CH + c) * HH + h) * WW + w];
        m = fmaxf(m, v[c]);
      }
      float s = 0.f;
      #pragma unroll
      for (int c = 0; c < CH; ++c) { v[c] = expf(v[c] - m); s += v[c]; }
      const float inv = 1.f / s;
      #pragma unroll
      for (int c = 0; c < CH; ++c) acc[c] = fmaf(v[c], inv, acc[c]);
    }
  }
  #pragma unroll
  for (int c = 0; c < CH; ++c)
    Qb[((b * CH + c) * PH + hp) * PW + wp] = acc[c] * (1.f / 16.f);
}

// ---------------------------------------------------------------------------
// Kernel 5: dual pac message: m{1,2}[b,c,p] = sum_j kern{1,2}[b,j,p]*Qb[b,c,p+off_j]
// (zero-padded Q patches). Hottest loop of the pipeline: nested dy/dx loops,
// branch-free clamped loads, kernels + Qb stay L2-resident (35MB << 192MB L2).
// ---------------------------------------------------------------------------
__global__ void pac_msg(const float* __restrict__ Qb,
                        const float* __restrict__ k1,
                        const float* __restrict__ k2,
                        float* __restrict__ m1o,
                        float* __restrict__ m2o) {
  int t = blockIdx.x * blockDim.x + threadIdx.x;
  if (t >= BN * CH * PH * PW) return;
  const int wp = t & 127, hp = (t >> 7) & 127;
  const int rest = t >> 14;
  const int c = rest % CH, b = rest / CH;

  const int kbase = b * K2 * (PH * PW) + hp * PW + wp;
  const int qbase = (b * CH + c) * (PH * PW);
  float m1 = 0.f, m2 = 0.f;
  int j = 0;
  for (int dy = -5; dy <= 5; ++dy) {
    const int yy = hp + dy;
    const bool rowok = (yy >= 0) && (yy < PH);
    const int yrow = qbase + min(max(yy, 0), PH - 1) * PW;
    #pragma unroll
    for (int dx = -5; dx <= 5; ++dx, ++j) {
      const int xx = wp + dx;
      const bool ok = rowok && (xx >= 0) && (xx < PW);
      const int xxc = min(max(xx, 0), PW - 1);
      float qv = Qb[yrow + xxc];          // always-valid address
      qv = ok ? qv : 0.f;
      m1 = fmaf(k1[kbase + j * (PH * PW)], qv, m1);
      m2 = fmaf(k2[kbase + j * (PH * PW)], qv, m2);
    }
  }
  const int o = ((b * CH + c) * PH + hp) * PW + wp;
  m1o[o] = m1;
  m2o[o] = m2;
}

// ---------------------------------------------------------------------------
// Kernel 6: bilinear x4 upsample (align_corners=False) of both messages +
// weighted combine with unary -> new logQ
// ---------------------------------------------------------------------------
__global__ void upsample_combine(const float* __restrict__ unary,
                                 const float* __restrict__ m1,
                                 const float* __restrict__ m2,
                                 const float* __restrict__ uw_p,
                                 const float* __restrict__ pw_p,
                                 float* __restrict__ out) {
  int t = blockIdx.x * blockDim.x + threadIdx.x;
  if (t >= BN * CH * HH * WW) return;
  const int w = t & 511, h = (t >> 9) & 511;
  const int rest = t >> 18;
  const int c = rest % CH, b = rest / CH;

  float sy = fminf(fmaxf((h + 0.5f) * 0.25f - 0.5f, 0.f), (float)(PH - 1));
  float sx = fminf(fmaxf((w + 0.5f) * 0.25f - 0.5f, 0.f), (float)(PW - 1));
  const int y0 = (int)sy, x0 = (int)sx;
  const int y1 = min(y0 + 1, PH - 1), x1 = min(x0 + 1, PW - 1);
  const float fy = sy - (float)y0, fx = sx - (float)x0;

  const int base = (b * CH + c) * (PH * PW);
  const int i00 = base + y0 * PW + x0, i01 = base + y0 * PW + x1;
  const int i10 = base + y1 * PW + x0, i11 = base + y1 * PW + x1;

  const float w00 = (1.f - fy) * (1.f - fx), w01 = (1.f - fy) * fx;
  const float w10 = fy * (1.f - fx),         w11 = fy * fx;

  const float b1 = m1[i00] * w00 + m1[i01] * w01 + m1[i10] * w10 + m1[i11] * w11;
  const float b2 = m2[i00] * w00 + m2[i01] * w01 + m2[i10] * w10 + m2[i11] * w11;

  out[t] = uw_p[0] * unary[t] + pw_p[0] * b1 + pw_p[1] * b2;
}

// ---------------------------------------------------------------------------
extern "C" void kernel_launch(void* const* d_in, const int* in_sizes, int n_in,
                              void* d_out, int out_size, void* d_ws, size_t ws_size,
                              hipStream_t stream) {
  (void)in_sizes; (void)n_in; (void)out_size; (void)ws_size;
  const float* x  = (const float*)d_in[0];  // [2,3,512,512]
  const float* wb = (const float*)d_in[1];  // [21,3,3,3] -> [21][27]
  const float* bb = (const float*)d_in[2];  // [21]
  const float* uw = (const float*)d_in[3];  // scalar
  const float* pw = (const float*)d_in[4];  // [2]
  float* out = (float*)d_out;               // [2,21,512,512]

  float* ws = (float*)d_ws;
  size_t off = 0;
  float* unary = ws + off; off += (size_t)BN * CH * HH * WW;   // 11,010,048
  float* featB = ws + off; off += (size_t)BN * 5 * PH * PW;    //    163,840
  float* featS = ws + off; off += (size_t)BN * 2 * PH * PW;    //     65,536
  float* kern1 = ws + off; off += (size_t)BN * K2 * PH * PW;   //  3,964,928
  float* kern2 = ws + off; off += (size_t)BN * K2 * PH * PW;   //  3,964,928
  float* Qb    = ws + off; off += (size_t)BN * CH * PH * PW;   //    688,128
  float* m1    = ws + off; off += (size_t)BN * CH * PH * PW;
  float* m2    = ws + off; off += (size_t)BN * CH * PH * PW;

  // backbone conv (WMMA GEMM): 2 * 512 * 32 * 2 = 65536 waves
  conv_unary_wmma<<<BN * HH * (WW / 16) * 2, 32, 0, stream>>>(x, wb, bb, unary);

  // edge features + Gaussian kernels (computed once)
  const int npool = BN * PH * PW;                  // 32768
  pooled_feats<<<npool / 256, 256, 0, stream>>>(x, featB, featS);
  gauss_kernel<5><<<npool / 256, 256, 0, stream>>>(featB, kern1);
  gauss_kernel<2><<<npool / 256, 256, 0, stream>>>(featS, kern2);

  // 5 mean-field steps; logQ starts as unary, then lives in d_out
  const int nmsg = BN * CH * PH * PW;              // 688,128
  const int nout = BN * CH * HH * WW;              // 11,010,048
  for (int it = 0; it < 5; ++it) {
    const float* lq = (it == 0) ? unary : out;
    softmax_pool<<<npool / 256, 256, 0, stream>>>(lq, Qb);
    pac_msg<<<nmsg / 256, 256, 0, stream>>>(Qb, kern1, kern2, m1, m2);
    upsample_combine<<<nout / 256, 256, 0, stream>>>(unary, m1, m2, uw, pw, out);
  }
}